// LinearPredictor_27376121545095
// MI455X (gfx1250) — compile-verified
//
#include <hip/hip_runtime.h>

typedef float v2f __attribute__((ext_vector_type(2)));
typedef float v8f __attribute__((ext_vector_type(8)));

#define LN_EPS 1e-5f
#define HF 128   // h_feats
// 2*HF = 256 = LayerNorm width

// ---------------------------------------------------------------------------
// Workspace layout (floats):
//   [0, n_nodes*4)            node stats float4 {rowsum, rowsumsq, dot_lo, dot_hi}
//   [n_nodes*4, +128*16)      G matrix, row-major [k][n]: col0=1, col1=gw_lo[k], col2=gw_hi[k]
//   [.., +2)                  scal[0]=S=sum(gw), scal[1]=C=sum(beta*W)+b
// ---------------------------------------------------------------------------

// Tiny prep: gw = gamma*W, build B-matrix G (128x16) and the two scalars.
__global__ __launch_bounds__(128) void prep_kernel(
    const float* __restrict__ gamma, const float* __restrict__ beta,
    const float* __restrict__ W, const float* __restrict__ bptr,
    float* __restrict__ G, float* __restrict__ scal)
{
    __shared__ float r1[128], r2[128];
    const int k = threadIdx.x;
    const float gwlo = gamma[k] * W[k];
    const float gwhi = gamma[HF + k] * W[HF + k];
    float* g = G + k * 16;
    g[0] = 1.0f; g[1] = gwlo; g[2] = gwhi;
#pragma unroll
    for (int i = 3; i < 16; ++i) g[i] = 0.0f;
    r1[k] = gwlo + gwhi;
    r2[k] = beta[k] * W[k] + beta[HF + k] * W[HF + k];
    __syncthreads();
    for (int off = 64; off > 0; off >>= 1) {
        if (k < off) { r1[k] += r1[k + off]; r2[k] += r2[k + off]; }
        __syncthreads();
    }
    if (k == 0) { scal[0] = r1[0]; scal[1] = r2[0] + bptr[0]; }
}

// Node-stats phase: one wave32 per 16-node tile.
// Chain 1: D1 = H_tile(16x128) x G(128x16)      -> col0=rowsum, col1=dot_lo, col2=dot_hi
// Chain 2: D2 = (H_tile .^2)   x ones_col       -> col0=rowsumsq
// using V_WMMA_F32_16X16X4_F32, K stepped by 4.
__global__ __launch_bounds__(32) void node_stats_kernel(
    const float* __restrict__ h, const float* __restrict__ G,
    float4* __restrict__ stats, int n_nodes)
{
    __shared__ float lds[2][16][16];
    const int lane = threadIdx.x;            // 0..31
    const int n    = lane & 15;              // column / row-in-tile index
    const int koff = (lane >> 4) << 1;       // 0 (lanes 0-15) or 2 (lanes 16-31)
    const int row0 = blockIdx.x * 16;

    int row = row0 + n;                      // A-matrix row this lane feeds
    if (row >= n_nodes) row = n_nodes - 1;   // uniform-shape clamp (no divergence)
    const float* __restrict__ hrow = h + (size_t)row * HF;

    // Constant B for chain 2: ones in column 0 only (same value in both K slots).
    const float b2e = (n == 0) ? 1.0f : 0.0f;
    v2f b2; b2.x = b2e; b2.y = b2e;

    v8f acc1 = {};
    v8f acc2 = {};

#pragma unroll 4
    for (int k0 = 0; k0 < HF; k0 += 4) {
        // A fragment: lane holds A[n][koff], A[n][koff+1] (16x4 f32 layout)
        v2f a = *(const v2f*)(hrow + k0 + koff);
        v2f asq; asq.x = a.x * a.x; asq.y = a.y * a.y;
        // B fragment: lane holds B[koff][n] (vgpr0) and B[koff+1][n] (vgpr1)
        const float* gp = G + (size_t)(k0 + koff) * 16 + n;
        v2f b1; b1.x = gp[0]; b1.y = gp[16];

        acc1 = __builtin_amdgcn_wmma_f32_16x16x4_f32(
            false, a,   false, b1, (short)0, acc1, false, false);
        acc2 = __builtin_amdgcn_wmma_f32_16x16x4_f32(
            false, asq, false, b2, (short)0, acc2, false, false);
    }

    // Spill D through LDS using the 16x16 f32 C/D layout:
    // VGPR v holds row v (lanes 0-15) / row v+8 (lanes 16-31), column = lane&15.
    const int half = lane >> 4;
#pragma unroll
    for (int v = 0; v < 8; ++v) {
        const int r = v + half * 8;
        lds[0][r][n] = acc1[v];
        lds[1][r][n] = acc2[v];
    }
    __syncthreads();

    if (lane < 16) {
        const int node = row0 + lane;
        if (node < n_nodes) {
            float4 o;
            o.x = lds[0][lane][0];   // rowsum
            o.y = lds[1][lane][0];   // rowsumsq
            o.z = lds[0][lane][1];   // dot(gw_lo, h[node])
            o.w = lds[0][lane][2];   // dot(gw_hi, h[node])
            stats[node] = o;
        }
    }
}

// Edge phase: 32 B of L2-resident stats per edge instead of 1 KB of h.
__global__ __launch_bounds__(256) void edge_score_kernel(
    const int* __restrict__ src, const int* __restrict__ dst,
    const float4* __restrict__ stats, const float* __restrict__ scal,
    float* __restrict__ out, int n_edges)
{
    const int e = blockIdx.x * blockDim.x + threadIdx.x;
    if (e >= n_edges) return;
    const float4 a = stats[src[e]];
    const float4 b = stats[dst[e]];
    const float S = scal[0];
    const float C = scal[1];
    const float inv = 1.0f / 256.0f;
    const float mu  = (a.x + b.x) * inv;
    const float var = (a.y + b.y) * inv - mu * mu;
    const float s3  = a.z + b.w;            // src uses gw_lo half, dst uses gw_hi half
    out[e] = (s3 - mu * S) * rsqrtf(var + LN_EPS) + C;
}

extern "C" void kernel_launch(void* const* d_in, const int* in_sizes, int n_in,
                              void* d_out, int out_size, void* d_ws, size_t ws_size,
                              hipStream_t stream) {
    const float* h     = (const float*)d_in[0];
    const int*   src   = (const int*)  d_in[1];
    const int*   dst   = (const int*)  d_in[2];
    const float* gamma = (const float*)d_in[3];
    const float* beta  = (const float*)d_in[4];
    const float* W     = (const float*)d_in[5];
    const float* b     = (const float*)d_in[6];

    const int n_nodes = in_sizes[0] / HF;
    const int n_edges = in_sizes[1];

    float*  ws    = (float*)d_ws;
    float4* stats = (float4*)ws;
    float*  G     = ws + (size_t)n_nodes * 4;
    float*  scal  = G + 128 * 16;
    float*  out   = (float*)d_out;

    prep_kernel<<<1, 128, 0, stream>>>(gamma, beta, W, b, G, scal);

    const int ntiles = (n_nodes + 15) / 16;
    node_stats_kernel<<<ntiles, 32, 0, stream>>>(h, G, stats, n_nodes);

    const int nblk = (n_edges + 255) / 256;
    edge_score_kernel<<<nblk, 256, 0, stream>>>(src, dst, stats, scal, out, n_edges);
}